// SpatialContextAwareTimeSeriesForecast_22136261443925
// MI455X (gfx1250) — compile-verified
//
#include <hip/hip_runtime.h>
#include <hip/hip_bf16.h>

typedef __attribute__((ext_vector_type(2))) float v2f;
typedef __attribute__((ext_vector_type(8))) float v8f;

#define BROWS 8192
#define EDIM  256
#define NSIM  2000
#define TOPK  10
#define NSTEP 9
#define UDIM  2000
#define IDIM  2000

__device__ __forceinline__ v8f wmma4(v2f a, v2f b, v8f c) {
  // D = A(16x4 f32) * B(4x16 f32) + C(16x16 f32)
  return __builtin_amdgcn_wmma_f32_16x16x4_f32(false, a, false, b, (short)0, c,
                                               false, false);
}

__device__ __forceinline__ float sigm(float x) {
  return 1.0f / (1.0f + __expf(-x));
}

// Stage W[:, n0:n0+16] transposed into lds: lds[c*ldss + k] = W[k*256 + n0 + c]
// 128-thread block, uniform trip count -> no EXEC divergence.
__device__ __forceinline__ void stage_wt(const float* __restrict__ W, int K,
                                         int n0, float* lds, int ldss) {
  int t = threadIdx.x;
  for (int i = t; i < 16 * K; i += 128) {
    int c = i & 15;
    int k = i >> 4;
    lds[c * ldss + k] = W[(size_t)k * EDIM + n0 + c];
  }
}

// ---------------------------------------------------------------- top-k + CNN
__global__ void topk_nei_kernel(const float* __restrict__ sim,
                                const int* __restrict__ ids,
                                const float* __restrict__ emb,
                                const float* __restrict__ cw,
                                const float* __restrict__ cb,
                                float* __restrict__ out) {
  __shared__ float vals[NSIM];
  __shared__ unsigned long long red[256];
  __shared__ int sel[TOPK];
  int b = blockIdx.x;
  int t = threadIdx.x;
  int row = ids[b];
  const float* src = sim + (size_t)row * NSIM;
  for (int i = t; i < NSIM; i += 256) vals[i] = src[i];
  __syncthreads();
  for (int k = 0; k < TOPK; ++k) {
    unsigned long long best = 0ULL;
    for (int i = t; i < NSIM; i += 256) {
      unsigned vb = __float_as_uint(vals[i]);
      vb ^= (0x80000000u | (unsigned)((int)vb >> 31));  // monotonic total order
      unsigned long long key =
          ((unsigned long long)vb << 32) | (unsigned)(~i);  // ties: lower idx
      if (key > best) best = key;
    }
    red[t] = best;
    __syncthreads();
    for (int s = 128; s > 0; s >>= 1) {
      if (t < s) {
        unsigned long long o = red[t + s];
        if (o > red[t]) red[t] = o;
      }
      __syncthreads();
    }
    if (t == 0) {
      int idx = (int)(~(unsigned)(red[0] & 0xFFFFFFFFULL));
      sel[k] = idx;
      vals[idx] = -1.0f;  // exclude
    }
    __syncthreads();
  }
  float acc = cb[0];
  int e = t;  // 256 threads == EDIM
  for (int k = 0; k < TOPK; ++k)
    acc += cw[k] * emb[(size_t)sel[k] * EDIM + e];
  out[(size_t)b * EDIM + e] = acc;
}

// ----------------------------------------------------------------- S builder
__global__ void make_s_kernel(const int* __restrict__ uid,
                              const int* __restrict__ iid,
                              const float* __restrict__ uemb,
                              const float* __restrict__ iemb,
                              const float* __restrict__ unei,
                              const float* __restrict__ inei,
                              float* __restrict__ S) {
  size_t i = (size_t)blockIdx.x * 256 + threadIdx.x;  // over B*256
  int b = (int)(i >> 8);
  int n = (int)(i & 255);
  float ue = uemb[(size_t)uid[b] * EDIM + n];
  float ie = iemb[(size_t)iid[b] * EDIM + n];
  float un = unei[i];
  float in_ = inei[i];
  size_t str = (size_t)BROWS * EDIM;
  S[0 * str + i] = ue + ie;
  S[1 * str + i] = ue + un;
  S[2 * str + i] = un + ie;
  S[3 * str + i] = un + in_;
}

// ------------------------------------------------ inter: gated dual-GEMM WMMA
__global__ void inter_kernel(const float* __restrict__ S,
                             const float* __restrict__ Wf,
                             const float* __restrict__ bf,
                             const float* __restrict__ Wi,
                             const float* __restrict__ bi,
                             float* __restrict__ cross) {
  extern __shared__ float lds[];
  const int ldss = EDIM + 4;  // 260: 4-bank stagger per column
  int t = threadIdx.x, lane = t & 31, wave = t >> 5;
  int hi = lane >> 4, col = lane & 15, koff = hi * 2;
  int j = blockIdx.y;
  const float* A = S + (size_t)j * BROWS * EDIM;
  int m0 = blockIdx.x * 64 + wave * 16;
  const float* Arow = A + (size_t)(m0 + col) * EDIM;
  float* ldsF = lds;
  float* ldsI = lds + 16 * ldss;
  for (int nt = 0; nt < 16; ++nt) {
    int n0 = nt * 16;
    __syncthreads();
    stage_wt(Wf, EDIM, n0, ldsF, ldss);
    stage_wt(Wi, EDIM, n0, ldsI, ldss);
    __syncthreads();
    v8f accF = {};
    v8f accI = {};
    const float* WcF = ldsF + col * ldss;
    const float* WcI = ldsI + col * ldss;
#pragma unroll 4
    for (int k0 = 0; k0 < EDIM; k0 += 4) {
      v2f a = *(const v2f*)(Arow + k0 + koff);
      v2f b0 = *(const v2f*)(WcF + k0 + koff);
      v2f b1 = *(const v2f*)(WcI + k0 + koff);
      accF = wmma4(a, b0, accF);
      accI = wmma4(a, b1, accI);
    }
    int n = n0 + col;
#pragma unroll
    for (int v = 0; v < 8; ++v) {
      int r = m0 + v + 8 * hi;
      float s = A[(size_t)r * EDIM + n];
      float f = sigm(accF[v] + bf[n]);
      float g = accI[v] + bi[n];
      cross[(size_t)r * 1024 + j * 256 + n] = f * g + (1.0f - f) * s;
    }
  }
}

// ----------------------------------------- generic GEMM + bias (+relu) WMMA
__global__ void gemm_bias_act_kernel(const float* __restrict__ A, int lda,
                                     int K, const float* __restrict__ W,
                                     const float* __restrict__ bias,
                                     float* __restrict__ out, int relu) {
  extern __shared__ float lds[];
  const int ldss = K + 4;
  int t = threadIdx.x, lane = t & 31, wave = t >> 5;
  int hi = lane >> 4, col = lane & 15, koff = hi * 2;
  int m0 = blockIdx.x * 64 + wave * 16;
  const float* Arow = A + (size_t)(m0 + col) * lda;
  for (int nt = 0; nt < 16; ++nt) {
    int n0 = nt * 16;
    __syncthreads();
    stage_wt(W, K, n0, lds, ldss);
    __syncthreads();
    v8f acc = {};
    const float* Wc = lds + col * ldss;
#pragma unroll 8
    for (int k0 = 0; k0 < K; k0 += 4) {
      v2f a = *(const v2f*)(Arow + k0 + koff);
      v2f b = *(const v2f*)(Wc + k0 + koff);
      acc = wmma4(a, b, acc);
    }
    int n = n0 + col;
#pragma unroll
    for (int v = 0; v < 8; ++v) {
      int r = m0 + v + 8 * hi;
      float o = acc[v] + bias[n];
      if (relu) o = o > 0.0f ? o : 0.0f;
      out[(size_t)r * EDIM + n] = o;
    }
  }
}

// ---------------------------------------------------------------- seq gather
__global__ void seq_kernel(const int* __restrict__ time_id,
                           const int* __restrict__ uid,
                           const int* __restrict__ iid,
                           const float* __restrict__ qos,
                           float* __restrict__ seq) {
  int i = blockIdx.x * 256 + threadIdx.x;  // over B*9
  if (i >= BROWS * NSTEP) return;
  int ts = i / BROWS;
  int b = i - ts * BROWS;
  size_t off =
      ((size_t)(time_id[b] + ts) * UDIM + uid[b]) * (size_t)IDIM + iid[b];
  seq[i] = qos[off];
}

// ------------------------------------------------- z/r gates: dual GEMM WMMA
__global__ void zr_kernel(const float* __restrict__ h,
                          const float* __restrict__ Wzh,
                          const float* __restrict__ Wrh,
                          const float* __restrict__ Wzx,
                          const float* __restrict__ Wrx,
                          const float* __restrict__ bz,
                          const float* __restrict__ br,
                          const float* __restrict__ cbuf,
                          const float* __restrict__ xrow,
                          float* __restrict__ zbuf, float* __restrict__ rbuf) {
  extern __shared__ float lds[];
  const int ldss = EDIM + 4;
  int t = threadIdx.x, lane = t & 31, wave = t >> 5;
  int hi = lane >> 4, col = lane & 15, koff = hi * 2;
  int m0 = blockIdx.x * 64 + wave * 16;
  const float* Arow = h + (size_t)(m0 + col) * EDIM;
  float* ldsZ = lds;
  float* ldsR = lds + 16 * ldss;
  for (int nt = 0; nt < 16; ++nt) {
    int n0 = nt * 16;
    __syncthreads();
    stage_wt(Wzh, EDIM, n0, ldsZ, ldss);
    stage_wt(Wrh, EDIM, n0, ldsR, ldss);
    __syncthreads();
    v8f accZ = {};
    v8f accR = {};
    const float* WcZ = ldsZ + col * ldss;
    const float* WcR = ldsR + col * ldss;
#pragma unroll 4
    for (int k0 = 0; k0 < EDIM; k0 += 4) {
      v2f a = *(const v2f*)(Arow + k0 + koff);
      v2f b0 = *(const v2f*)(WcZ + k0 + koff);
      v2f b1 = *(const v2f*)(WcR + k0 + koff);
      accZ = wmma4(a, b0, accZ);
      accR = wmma4(a, b1, accR);
    }
    int n = n0 + col;
    float wzx = Wzx[n], wrx = Wrx[n], bzn = bz[n], brn = br[n];
#pragma unroll
    for (int v = 0; v < 8; ++v) {
      int r = m0 + v + 8 * hi;
      float x = xrow[r];
      float cv = cbuf[(size_t)r * EDIM + n];
      zbuf[(size_t)r * EDIM + n] = sigm(accZ[v] + x * wzx + bzn + cv);
      rbuf[(size_t)r * EDIM + n] = sigm(accR[v] + x * wrx + brn + cv);
    }
  }
}

// ------------------------------------------------ ht = tanh((h*r)@Whh + ...)
__global__ void ht_kernel(const float* __restrict__ h,
                          const float* __restrict__ rbuf,
                          const float* __restrict__ Whh,
                          const float* __restrict__ Whx,
                          const float* __restrict__ bh,
                          const float* __restrict__ cbuf,
                          const float* __restrict__ xrow,
                          float* __restrict__ htbuf) {
  extern __shared__ float lds[];
  const int ldss = EDIM + 4;
  int t = threadIdx.x, lane = t & 31, wave = t >> 5;
  int hi = lane >> 4, col = lane & 15, koff = hi * 2;
  int m0 = blockIdx.x * 64 + wave * 16;
  const float* Hrow = h + (size_t)(m0 + col) * EDIM;
  const float* Rrow = rbuf + (size_t)(m0 + col) * EDIM;
  for (int nt = 0; nt < 16; ++nt) {
    int n0 = nt * 16;
    __syncthreads();
    stage_wt(Whh, EDIM, n0, lds, ldss);
    __syncthreads();
    v8f acc = {};
    const float* Wc = lds + col * ldss;
#pragma unroll 4
    for (int k0 = 0; k0 < EDIM; k0 += 4) {
      v2f a0 = *(const v2f*)(Hrow + k0 + koff);
      v2f a1 = *(const v2f*)(Rrow + k0 + koff);
      v2f a = a0 * a1;  // elementwise h * r feeds the A operand
      v2f b = *(const v2f*)(Wc + k0 + koff);
      acc = wmma4(a, b, acc);
    }
    int n = n0 + col;
    float whx = Whx[n], bhn = bh[n];
#pragma unroll
    for (int v = 0; v < 8; ++v) {
      int r = m0 + v + 8 * hi;
      float x = xrow[r];
      float cv = cbuf[(size_t)r * EDIM + n];
      htbuf[(size_t)r * EDIM + n] = tanhf(acc[v] + x * whx + bhn + cv);
    }
  }
}

// ---------------------------------------------------------- h update (elemwise)
__global__ void combine_kernel(float* __restrict__ h,
                               const float* __restrict__ z,
                               const float* __restrict__ ht) {
  size_t i = (size_t)blockIdx.x * 256 + threadIdx.x;
  float zz = z[i];
  h[i] = (1.0f - zz) * h[i] + zz * ht[i];
}

// ------------------------------------------------------------- final project
__global__ void out_kernel(const float* __restrict__ h,
                           const float* __restrict__ Wp,
                           const float* __restrict__ bp,
                           float* __restrict__ y) {
  int t = threadIdx.x;
  int lane = t & 31;
  int row = blockIdx.x * 8 + (t >> 5);
  float acc = 0.0f;
#pragma unroll
  for (int i = 0; i < 8; ++i) {
    int e = lane + i * 32;
    float v = h[(size_t)row * EDIM + e];
    v = v > 0.0f ? v : 0.0f;
    acc += v * Wp[e];
  }
  for (int off = 16; off > 0; off >>= 1) acc += __shfl_down(acc, off, 32);
  if (lane == 0) y[row] = acc + bp[0];
}

extern "C" void kernel_launch(void* const* d_in, const int* in_sizes, int n_in,
                              void* d_out, int out_size, void* d_ws,
                              size_t ws_size, hipStream_t stream) {
  const int* time_id = (const int*)d_in[0];
  const int* user_id = (const int*)d_in[1];
  const int* item_id = (const int*)d_in[2];
  const float* uu_sim = (const float*)d_in[3];
  const float* ii_sim = (const float*)d_in[4];
  const float* qos = (const float*)d_in[5];
  const float* user_emb = (const float*)d_in[6];
  const float* item_emb = (const float*)d_in[7];
  const float* ucnn_w = (const float*)d_in[8];
  const float* ucnn_b = (const float*)d_in[9];
  const float* icnn_w = (const float*)d_in[10];
  const float* icnn_b = (const float*)d_in[11];
  const float* Wi = (const float*)d_in[12];
  const float* bi = (const float*)d_in[13];
  const float* Wf = (const float*)d_in[14];
  const float* bf = (const float*)d_in[15];
  const float* Wfc = (const float*)d_in[16];
  const float* bfc = (const float*)d_in[17];
  const float* Wzh = (const float*)d_in[18];
  const float* Wzx = (const float*)d_in[19];
  const float* bz = (const float*)d_in[20];
  const float* Wrh = (const float*)d_in[21];
  const float* Wrx = (const float*)d_in[22];
  const float* br = (const float*)d_in[23];
  const float* Whh = (const float*)d_in[24];
  const float* Whx = (const float*)d_in[25];
  const float* bh = (const float*)d_in[26];
  const float* Wch = (const float*)d_in[27];
  const float* bc = (const float*)d_in[28];
  const float* Wp = (const float*)d_in[29];
  const float* bp = (const float*)d_in[30];

  const size_t BE = (size_t)BROWS * EDIM;
  float* ws = (float*)d_ws;
  float* u_nei = ws;             // B*256
  float* i_nei = u_nei + BE;     // B*256
  float* S = i_nei + BE;         // 4*B*256
  float* cross = S + 4 * BE;     // B*1024
  float* hbuf = cross + 4 * BE;  // B*256
  float* cbuf = hbuf + BE;       // B*256
  float* seq = cbuf + BE;        // 9*B
  float* zbuf = S;               // S is dead after cross -> reuse
  float* rbuf = S + BE;
  float* htb = S + 2 * BE;

  const int ldssE = EDIM + 4;
  const size_t shm1 = (size_t)16 * ldssE * sizeof(float);
  const size_t shm2 = 2 * shm1;
  const size_t shmFC = (size_t)16 * (1024 + 4) * sizeof(float);

  // neighbor CNN features via top-k
  topk_nei_kernel<<<BROWS, 256, 0, stream>>>(uu_sim, user_id, user_emb, ucnn_w,
                                             ucnn_b, u_nei);
  topk_nei_kernel<<<BROWS, 256, 0, stream>>>(ii_sim, item_id, item_emb, icnn_w,
                                             icnn_b, i_nei);
  // the four pairwise sums
  make_s_kernel<<<BROWS, 256, 0, stream>>>(user_id, item_id, user_emb, item_emb,
                                           u_nei, i_nei, S);
  // gated interactions -> cross (B x 1024)
  inter_kernel<<<dim3(BROWS / 64, 4), 128, shm2, stream>>>(S, Wf, bf, Wi, bi,
                                                           cross);
  // h0 = cross @ Wfc + bfc
  gemm_bias_act_kernel<<<BROWS / 64, 128, shmFC, stream>>>(cross, 1024, 1024,
                                                           Wfc, bfc, hbuf, 0);
  // c = relu(h0 @ Wch + bc)
  gemm_bias_act_kernel<<<BROWS / 64, 128, shm1, stream>>>(hbuf, EDIM, EDIM, Wch,
                                                          bc, cbuf, 1);
  // qos sequence gather
  seq_kernel<<<(BROWS * NSTEP + 255) / 256, 256, 0, stream>>>(time_id, user_id,
                                                              item_id, qos, seq);
  // GRU-style scan, 9 steps
  for (int ts = 0; ts < NSTEP; ++ts) {
    const float* xrow = seq + (size_t)ts * BROWS;
    zr_kernel<<<BROWS / 64, 128, shm2, stream>>>(hbuf, Wzh, Wrh, Wzx, Wrx, bz,
                                                 br, cbuf, xrow, zbuf, rbuf);
    ht_kernel<<<BROWS / 64, 128, shm1, stream>>>(hbuf, rbuf, Whh, Whx, bh, cbuf,
                                                 xrow, htb);
    combine_kernel<<<BROWS, 256, 0, stream>>>(hbuf, zbuf, htb);
  }
  // y = relu(h) @ Wp + bp
  out_kernel<<<BROWS / 8, 256, 0, stream>>>(hbuf, Wp, bp, (float*)d_out);
}